// FeatuesPoints_76905684402437
// MI455X (gfx1250) — compile-verified
//
#include <hip/hip_runtime.h>

// Problem geometry
#define NIMG 16
#define IMGH 1024
#define IMGW 1024
#define TILE 128          // output tile per workgroup
#define HALO 17           // 4 rounds * (3 conv + 1 pool) + 1 avgpool
#define EXT  162          // TILE + 2*HALO (valid field extent)
#define S    162          // LDS row stride in floats (even -> 8B-aligned b64 loads)
#define ROWS 168          // EXT + 6 zeroed guard rows (WMMA overshoot reads stay finite)
#define NTHREADS 256      // 8 wave32 waves

typedef float v2f __attribute__((ext_vector_type(2)));
typedef float v8f __attribute__((ext_vector_type(8)));
typedef unsigned int v4u __attribute__((ext_vector_type(4)));
typedef int v4i __attribute__((ext_vector_type(4)));
typedef int v8i __attribute__((ext_vector_type(8)));

#ifndef __has_builtin
#define __has_builtin(x) 0
#endif
#if __has_builtin(__builtin_amdgcn_tensor_load_to_lds) && __has_builtin(__builtin_amdgcn_s_wait_tensorcnt)
#define HAVE_TDM 1
#else
#define HAVE_TDM 0
#endif
#if __has_builtin(__builtin_amdgcn_sched_barrier)
#define SCHED_FENCE() __builtin_amdgcn_sched_barrier(0)
#else
#define SCHED_FENCE()
#endif

__device__ __forceinline__ int clampi(int v, int lo, int hi) {
    return v < lo ? lo : (v > hi ? hi : v);
}

// LDS (218 KB) already limits us to 1 workgroup / WGP (2 waves per SIMD);
// tell the register allocator so the 84-VGPR B-fragment array doesn't spill.
__global__ void __launch_bounds__(NTHREADS, 1)
featpts_fused(const float* __restrict__ x,
              const float* __restrict__ kern,
              float* __restrict__ out,
              long long N)
{
    extern __shared__ float lds[];
    float* bufA = lds;                    // ROWS*S: current field (+ guard rows)
    float* bufB = lds + ROWS * S;         // ROWS*S: conv scratch
    float* wtab = lds + 2 * ROWS * S;     // 49 floats: 7x7 kernel

    const int tid  = threadIdx.x;
    const int lane = tid & 31;
    const int wave = tid >> 5;

    const int img    = blockIdx.x >> 6;   // 64 tiles per image
    const int tileId = blockIdx.x & 63;
    const int tty = tileId >> 3, ttx = tileId & 7;
    const int gy0 = tty * TILE;
    const int gx0 = ttx * TILE;
    const float* src = x + (long long)img * IMGH * IMGW;
    // tile whose full 162x162 halo'd field lies inside the image
    const bool interior = (tty >= 1) && (tty <= 6) && (ttx >= 1) && (ttx <= 6);

    if (tid < 49) wtab[tid] = kern[tid];

    // ---- Stage 0: fill bufA (162x162 field + 6 guard rows of zeros) ----
    if (interior) {
        for (int idx = tid; idx < (ROWS - EXT) * S; idx += NTHREADS)
            bufA[EXT * S + idx] = 0.f;
#if HAVE_TDM
        if (wave == 0) {
            // 2-D TDM descriptor: tile 162x162 of 4B elements, row stride 1024,
            // source = &src[(gy0-17)*1024 + gx0-17], dest = LDS offset 0 (bufA).
            unsigned long long ga = (unsigned long long)(uintptr_t)
                (src + (size_t)(gy0 - HALO) * IMGW + (gx0 - HALO));
            v4u g0;
            g0[0] = 1u;                                   // count=1, valid, no gather
            g0[1] = 0u;                                   // lds_addr = 0 (bufA)
            g0[2] = (unsigned)(ga & 0xFFFFFFFFu);         // global_addr[31:0]
            g0[3] = (unsigned)((ga >> 32) & 0x01FFFFFFu)  // global_addr[56:32]
                    | (2u << 30);                         // type = 2 ("image")
            v8i g1;
            g1[0] = (int)(2u << 16);                      // data_size = 4B; no mask/pad
            g1[1] = 0;                                    // tensor_dim0[15:0]=0 (dim0 = 1<<20)
            g1[2] = 0x0010;                               // tensor_dim0[31:16]; tensor_dim1 low = 0
            g1[3] = (int)((162u << 16) | 0x0010u);        // tensor_dim1 hi; tile_dim0 = 162
            g1[4] = 162;                                  // tile_dim1 = 162; tile_dim2 = 0
            g1[5] = 1024;                                 // tensor_dim0_stride = 1024
            g1[6] = 0;                                    // stride0 hi; dim1_stride low16 = 0
            g1[7] = 16;                                   // tensor_dim1_stride = 1<<20
            v4i gz = {0, 0, 0, 0};
#if __clang_major__ >= 23
            v8i gz8 = {0, 0, 0, 0, 0, 0, 0, 0};
            __builtin_amdgcn_tensor_load_to_lds(g0, g1, gz, gz, gz8, 0);
#else
            __builtin_amdgcn_tensor_load_to_lds(g0, g1, gz, gz, 0);
#endif
            __builtin_amdgcn_s_wait_tensorcnt((short)0);
        }
#else
        for (int idx = tid; idx < EXT * EXT; idx += NTHREADS) {
            int i = idx / EXT, j = idx - i * EXT;
            bufA[i * S + j] = src[(size_t)(gy0 + i - HALO) * IMGW + (gx0 + j - HALO)];
        }
#endif
    } else {
        // border tile: checked scalar fill, zero padding outside the image
        for (int idx = tid; idx < ROWS * S; idx += NTHREADS) {
            int i = idx / S, j = idx - i * S;
            float v = 0.f;
            if (i < EXT && j < EXT) {
                int gy = gy0 + i - HALO, gx = gx0 + j - HALO;
                if (gy >= 0 && gy < IMGH && gx >= 0 && gx < IMGW)
                    v = src[gy * IMGW + gx];
            }
            bufA[i * S + j] = v;
        }
    }
    __syncthreads();

    const int m     = lane & 15;   // A-row / B-col / D-col index
    const int khalf = lane >> 4;   // which K pair this half-wave holds

    // ---- Hoisted B fragments: banded weights W[c][n] = w[dy][c-n], n == m ----
    v2f bf[7][6];
    #pragma unroll
    for (int dy = 0; dy < 7; ++dy)
        #pragma unroll
        for (int kb = 0; kb < 6; ++kb) {
            int c0 = kb * 4 + khalf * 2;
            int d0 = c0 - m, d1 = d0 + 1;
            float w0 = wtab[dy * 7 + clampi(d0, 0, 6)];
            float w1 = wtab[dy * 7 + clampi(d1, 0, 6)];
            bf[dy][kb].x = ((unsigned)d0 <= 6u) ? w0 : 0.f;
            bf[dy][kb].y = ((unsigned)d1 <= 6u) ? w1 : 0.f;
        }

    const int ty16 = tid >> 4, tx16 = tid & 15;   // 16x16 thread grid for pooling

    // ---- 4 rounds of conv7 (WMMA) + maxpool3 ----
    for (int r = 0; r < 4; ++r) {
        const int lo = 4 * r;
        const int hi = EXT - 4 * r;

        // conv: bufA -> bufB over valid output region [lo+3, hi-3)
        const int o0  = lo + 3;
        const int cw  = (hi - 3) - o0;
        const int NSX = (cw + 15) >> 4;      // 10,10,9,9 -> pairs-per-row is always 5
        const int nPairs = NSX * 5;
        for (int p = wave; p < nPairs; p += 8) {   // wave-uniform: EXEC all-1s
            const int sy  = p / 5;                 // constant divisor -> mulhi/shift
            const int sxp = p - sy * 5;
            const int sx0 = 2 * sxp;
            const bool have1 = (sx0 + 1 < NSX);    // tail: compute 2nd subtile, skip stores
            const int oy  = o0 + sy * 16;
            const int ox0 = o0 + sx0 * 16;
            // shared 10-block window: chain0 uses blocks 0..5, chain1 (at ox0+16) 4..9
            const float* base = bufA + (oy + m - 3) * S + (ox0 - 3 + 2 * khalf);
            v8f acc0 = {0.f, 0.f, 0.f, 0.f, 0.f, 0.f, 0.f, 0.f};
            v8f acc1 = {0.f, 0.f, 0.f, 0.f, 0.f, 0.f, 0.f, 0.f};
            v2f cur[10], nxt[10];
            #pragma unroll
            for (int q = 0; q < 10; ++q)
                cur[q] = *(const v2f*)(base + q * 4);
            #pragma unroll
            for (int dy = 0; dy < 7; ++dy) {
                if (dy < 6) {
                    #pragma unroll
                    for (int q = 0; q < 10; ++q)
                        nxt[q] = *(const v2f*)(base + (dy + 1) * S + q * 4);
                }
                SCHED_FENCE();   // keep row dy+1 loads above row dy's WMMAs
                #pragma unroll
                for (int kb = 0; kb < 6; ++kb) {
                    acc0 = __builtin_amdgcn_wmma_f32_16x16x4_f32(
                        false, cur[kb],     false, bf[dy][kb], (short)0, acc0, false, false);
                    acc1 = __builtin_amdgcn_wmma_f32_16x16x4_f32(
                        false, cur[kb + 4], false, bf[dy][kb], (short)0, acc1, false, false);
                }
                #pragma unroll
                for (int q = 0; q < 10; ++q) cur[q] = nxt[q];
            }
            #pragma unroll
            for (int v = 0; v < 8; ++v)
                bufB[(oy + v + 8 * khalf) * S + ox0 + m] = acc0[v];
            if (have1) {
                #pragma unroll
                for (int v = 0; v < 8; ++v)
                    bufB[(oy + v + 8 * khalf) * S + ox0 + 16 + m] = acc1[v];
            }
        }
        __syncthreads();

        // maxpool3: bufB -> bufA over [lo+4, hi-4)
        const int p0 = lo + 4;
        const int p1 = hi - 4;
        if (interior) {
            for (int i = p0 + ty16; i < p1; i += 16)
                for (int j = p0 + tx16; j < p1; j += 16) {
                    const float* c = bufB + i * S + j;
                    float mv = c[-S - 1];
                    mv = c[-S]     > mv ? c[-S]     : mv;
                    mv = c[-S + 1] > mv ? c[-S + 1] : mv;
                    mv = c[-1]     > mv ? c[-1]     : mv;
                    mv = c[0]      > mv ? c[0]      : mv;
                    mv = c[1]      > mv ? c[1]      : mv;
                    mv = c[S - 1]  > mv ? c[S - 1]  : mv;
                    mv = c[S]      > mv ? c[S]      : mv;
                    mv = c[S + 1]  > mv ? c[S + 1]  : mv;
                    bufA[i * S + j] = mv;
                }
        } else {
            // image border: maxpool pads -inf (skip out-of-image taps);
            // out-of-image positions become 0 (zero padding for the next conv)
            for (int i = p0 + ty16; i < p1; i += 16)
                for (int j = p0 + tx16; j < p1; j += 16) {
                    int gy = gy0 + i - HALO, gx = gx0 + j - HALO;
                    float mv = 0.f;
                    if (gy >= 0 && gy < IMGH && gx >= 0 && gx < IMGW) {
                        mv = -3.402823466e38f;
                        #pragma unroll
                        for (int di = -1; di <= 1; ++di)
                            #pragma unroll
                            for (int dj = -1; dj <= 1; ++dj) {
                                int yy = gy + di, xx = gx + dj;
                                if (yy >= 0 && yy < IMGH && xx >= 0 && xx < IMGW) {
                                    float t = bufB[(i + di) * S + (j + dj)];
                                    mv = t > mv ? t : mv;
                                }
                            }
                    }
                    bufA[i * S + j] = mv;
                }
        }
        __syncthreads();
    }

    // ---- Final: (>0.5) -> avgpool3 (/9 always) -> (>0.8) -> y, mask ----
    float* dsty = out;
    float* dstm = out + N;
    for (int i = ty16; i < TILE; i += 16)
        for (int j = tx16; j < TILE; j += 16) {
            const float* c = bufA + (HALO + i) * S + (HALO + j);
            int cnt = 0;
            cnt += c[-S - 1] > 0.5f;  cnt += c[-S] > 0.5f;  cnt += c[-S + 1] > 0.5f;
            cnt += c[-1]     > 0.5f;  cnt += c[0]  > 0.5f;  cnt += c[1]      > 0.5f;
            cnt += c[S - 1]  > 0.5f;  cnt += c[S]  > 0.5f;  cnt += c[S + 1]  > 0.5f;
            float yv = (cnt >= 8) ? 1.f : 0.f;   // cnt/9 > 0.8  <=>  cnt >= 8
            long long o = (long long)img * IMGH * IMGW
                        + (long long)(gy0 + i) * IMGW + (gx0 + j);
            dsty[o] = yv;   // y
            dstm[o] = yv;   // mask = (y > 0.5), identical since y in {0,1}
        }
}

extern "C" void kernel_launch(void* const* d_in, const int* in_sizes, int n_in,
                              void* d_out, int out_size, void* d_ws, size_t ws_size,
                              hipStream_t stream)
{
    (void)in_sizes; (void)n_in; (void)out_size; (void)d_ws; (void)ws_size;
    const float* x = (const float*)d_in[0];
    const float* k = (const float*)d_in[1];
    float* out = (float*)d_out;

    const long long N = (long long)NIMG * IMGH * IMGW;
    const size_t smem = (size_t)(2 * ROWS * S + 64) * sizeof(float);  // ~218 KB of 320 KB WGP LDS

    hipFuncSetAttribute(reinterpret_cast<const void*>(featpts_fused),
                        hipFuncAttributeMaxDynamicSharedMemorySize, (int)smem);

    dim3 grid(NIMG * 64);     // 16 images * 8*8 tiles
    dim3 block(NTHREADS);
    featpts_fused<<<grid, block, smem, stream>>>(x, k, out, N);
}